// GaussianIntegral2D_49100066128336
// MI455X (gfx1250) — compile-verified
//
#include <hip/hip_runtime.h>

typedef __attribute__((ext_vector_type(2))) float v2f;
typedef __attribute__((ext_vector_type(4))) float v4f;
typedef __attribute__((ext_vector_type(8))) float v8f;

#define LOG2PI 1.8378770664093453f
#define QPTS 49

// ---------------------------------------------------------------------------
// Kernel 1: gauss_pts [N,49,2] — the 205 MB output. One thread produces two
// consecutive (x,y) points and issues a single 16-byte coalesced store
// (global_store_b128, 512B contiguous per wave). Rot/trans reads hit L0
// (49 points share one cell's 6 floats).
// ---------------------------------------------------------------------------
__global__ __launch_bounds__(256) void gauss_pts_kernel(
    const float* __restrict__ rot, const float* __restrict__ trans,
    const float* __restrict__ eta, float* __restrict__ out_pts, int npairs)
{
    int i = blockIdx.x * 256 + threadIdx.x;   // pair index
    if (i >= npairs) return;
    unsigned p0 = 2u * (unsigned)i;

    float vals[4];
#pragma unroll
    for (int j = 0; j < 2; ++j) {
        unsigned p    = p0 + (unsigned)j;
        unsigned cell = p / 49u;              // magic-multiply, no div unit
        unsigned q    = p - 49u * cell;
        float ex = eta[q];                    // eta_01[0, q]
        float ey = eta[QPTS + q];             // eta_01[1, q]
        const float* R = rot   + 4u * cell;
        const float* T = trans + 2u * cell;
        vals[2 * j + 0] = fmaf(R[0], ex, fmaf(R[1], ey, T[0]));
        vals[2 * j + 1] = fmaf(R[2], ex, fmaf(R[3], ey, T[1]));
    }
    v4f v = {vals[0], vals[1], vals[2], vals[3]};
    *(v4f*)(out_pts + 2u * p0) = v;           // p0 even -> 16B aligned
}

// ---------------------------------------------------------------------------
// Kernel 2: integral [N]. One wave = 16 cells. The 49-term weighted sum is a
// [16x52]x[52x16] f32 matmul done as 13 chained V_WMMA_F32_16X16X4_F32 ops
// (weights replicated across the 16 columns; K padded with zero weights).
// A-layout (ISA 7.12.2, 32-bit A 16x4): lane m=L&15 is row M; lane-half
// selects K-pair (half0 -> K0,K1 ; half1 -> K2,K3), i.e. this lane supplies
// e(q0), e(q0+1) with q0 = 4c + 2*half. B uses the matching K striping.
// ---------------------------------------------------------------------------
__global__ __launch_bounds__(256) void integral_kernel(
    const float* __restrict__ means, const float* __restrict__ covars,
    const float* __restrict__ rot,   const float* __restrict__ trans,
    const float* __restrict__ eta,   const float* __restrict__ wts,
    float* __restrict__ out_int)
{
    __shared__ float sw[52], se0[52], se1[52];
    int tid = threadIdx.x;
    if (tid < 52) {
        bool ok = tid < QPTS;
        sw[tid]  = ok ? wts[tid]        : 0.0f;   // zero-pad K=49..51
        se0[tid] = ok ? eta[tid]        : 0.0f;
        se1[tid] = ok ? eta[QPTS + tid] : 0.0f;
    }
    __syncthreads();

    int lane = tid & 31;
    int wave = tid >> 5;
    int m    = lane & 15;        // WMMA row M  -> cell within wave tile
    int half = lane >> 4;        // K-pair selector
    unsigned tile = (unsigned)blockIdx.x * 128u + (unsigned)wave * 16u;
    unsigned cell = tile + (unsigned)m;

    // per-cell parameters (lanes L and L+16 duplicate -> L0 cache hits)
    float mx = means[2u * cell], my = means[2u * cell + 1u];
    const float* C = covars + 4u * cell;
    float c00 = C[0], c01 = C[1], c10 = C[2], c11 = C[3];
    const float* R = rot + 4u * cell;
    float r00 = R[0], r01 = R[1], r10 = R[2], r11 = R[3];
    float tx = trans[2u * cell], ty = trans[2u * cell + 1u];

    float det     = fmaf(c00, c11, -c01 * c10);
    float inv_det = 1.0f / det;
    float lnorm   = fmaf(-0.5f, __logf(det), -LOG2PI);
    float csum    = c01 + c10;

    v8f acc = {};
#pragma unroll
    for (int c = 0; c < 13; ++c) {
        int q0 = 4 * c + 2 * half;
        float e[2], b[2];
#pragma unroll
        for (int j = 0; j < 2; ++j) {
            int q = q0 + j;
            float ex = se0[q], ey = se1[q];
            float px = fmaf(r00, ex, fmaf(r01, ey, tx));
            float py = fmaf(r10, ex, fmaf(r11, ey, ty));
            float dx = px - mx, dy = py - my;
            float quad = (c11 * dx * dx - csum * dx * dy + c00 * dy * dy) * inv_det;
            e[j] = __expf(fmaf(-0.5f, quad, lnorm));   // w=0 kills q>=49 terms
            b[j] = sw[q];
        }
        v2f a2 = {e[0], e[1]};
        v2f b2 = {b[0], b[1]};
        // D(16x16) += A(16x4) x B(4x16); every column of D equals the integral
        acc = __builtin_amdgcn_wmma_f32_16x16x4_f32(
            /*neg_a=*/false, a2, /*neg_b=*/false, b2,
            /*c_mod=*/(short)0, acc, /*reuse_a=*/false, /*reuse_b=*/false);
    }

    // C/D layout: lane L, VGPR r -> D[r + 8*(L>>4)][L&15]. Column 0 suffices:
    // lane 0 holds cells tile+0..7, lane 16 holds tile+8..15.
    if (m == 0) {
        unsigned base = tile + 8u * (unsigned)half;
#pragma unroll
        for (int r = 0; r < 8; ++r) {
            float v = acc[r];
            v = fminf(fmaxf(v, 0.0f), 1.0f);
            out_int[base + (unsigned)r] = v;
        }
    }
}

// ---------------------------------------------------------------------------
extern "C" void kernel_launch(void* const* d_in, const int* in_sizes, int n_in,
                              void* d_out, int out_size, void* d_ws, size_t ws_size,
                              hipStream_t stream) {
    const float* means  = (const float*)d_in[0];  // [B1,B2,2]
    const float* covars = (const float*)d_in[1];  // [B1,B2,2,2]
    const float* rot    = (const float*)d_in[2];  // [B1,B2,2,2]
    const float* trans  = (const float*)d_in[3];  // [B1,B2,2]
    const float* eta    = (const float*)d_in[4];  // [2,49]
    const float* wts    = (const float*)d_in[5];  // [49]

    const int N = in_sizes[0] / 2;                // 524288 cells
    float* out_pts = (float*)d_out;               // [N,49,2]
    float* out_int = (float*)d_out + (size_t)N * QPTS * 2;  // [N]

    int npairs = N * QPTS / 2;                    // N*49 is even (N even)
    gauss_pts_kernel<<<(npairs + 255) / 256, 256, 0, stream>>>(
        rot, trans, eta, out_pts, npairs);

    // 128 cells per 256-thread block (8 waves x 16 cells)
    integral_kernel<<<N / 128, 256, 0, stream>>>(
        means, covars, rot, trans, eta, wts, out_int);
}